// WindowCCC_17557826306351
// MI455X (gfx1250) — compile-verified
//
#include <hip/hip_runtime.h>
#include <hip/hip_bf16.h>

// WindowCCC on MI455X (gfx1250, wave32).
// Bandwidth-bound (131 MB @ 23.3 TB/s ~ 5.6us). One wave per 500-elem window.
// Window totals accumulated via V_WMMA_F32_16X16X4_F32 with all-ones A:
// D[m,n] = sum_k B[k,n] (+C), rows identical -> 0.5 * sum_lanes(acc[0]) is the
// exact f32 total of every B element fed in, independent of placement. So we
// load float4/lane (global_load_b128), split into two 4x16 B tiles, and feed
// 128 elements per WMMA pair; the 116-elem tail is zero-padded (exact for sums).
// Suffix sums at shift n = total - prefix(first n); prefix scan covers only the
// first 128 elements (shifts 0..124).

#define WIN     500
#define NSHIFT  125
#define INV_WM1 (1.0f / 499.0f)

typedef __attribute__((ext_vector_type(2))) float v2f;
typedef __attribute__((ext_vector_type(8))) float v8f;

#define WMMA_ACC(acc, b) \
    acc = __builtin_amdgcn_wmma_f32_16x16x4_f32(false, ones, false, (b), (short)0, acc, false, false)

static __device__ __forceinline__ float wave_sum(float v) {
#pragma unroll
    for (int off = 16; off > 0; off >>= 1)
        v += __shfl_xor(v, off, 32);
    return v;
}

static __device__ __forceinline__ float wave_min(float v) {
#pragma unroll
    for (int off = 16; off > 0; off >>= 1)
        v = fminf(v, __shfl_xor(v, off, 32));
    return v;
}

// exclusive scan across the 32 lanes of a wave
static __device__ __forceinline__ float wave_exscan(float v, int lane) {
    float s = v;
#pragma unroll
    for (int off = 1; off < 32; off <<= 1) {
        float n = __shfl_up(s, off, 32);
        if (lane >= off) s += n;
    }
    return s - v;
}

__global__ __launch_bounds__(256) void window_ccc_kernel(
    const float* __restrict__ pred, const float* __restrict__ gt,
    float* __restrict__ win_min, int num_wins) {

    const int lane = threadIdx.x & 31;
    const int wave = (int)((blockIdx.x * blockDim.x + threadIdx.x) >> 5);
    if (wave >= num_wins) return;  // uniform per wave -> EXEC stays full

    const float* p = pred + (long)wave * WIN;
    const float* g = gt   + (long)wave * WIN;

    __builtin_prefetch(p, 0, 3);
    __builtin_prefetch(g, 0, 3);

    // ---- Phase 1: window totals via f32 WMMA (A = ones => column sums) ----
    v2f ones; ones[0] = 1.0f; ones[1] = 1.0f;
    v8f acc_p  = {};
    v8f acc_pp = {};
    v8f acc_g  = {};
    v8f acc_gg = {};
    v8f acc_pg = {};

#pragma unroll
    for (int c = 0; c < 4; ++c) {               // 4 x 128 elems (last zero-padded)
        const int off = c * 128 + lane * 4;
        float4 pv = make_float4(0.f, 0.f, 0.f, 0.f);
        float4 gv = make_float4(0.f, 0.f, 0.f, 0.f);
        if (off + 3 < WIN) {                    // c<3: always; c==3: lanes 0..28
            pv = *(const float4*)(p + off);     // global_load_b128, coalesced
            gv = *(const float4*)(g + off);
        }
        v2f bp0; bp0[0] = pv.x; bp0[1] = pv.y;
        v2f bp1; bp1[0] = pv.z; bp1[1] = pv.w;
        v2f bg0; bg0[0] = gv.x; bg0[1] = gv.y;
        v2f bg1; bg1[0] = gv.z; bg1[1] = gv.w;
        WMMA_ACC(acc_p,  bp0);        WMMA_ACC(acc_p,  bp1);
        WMMA_ACC(acc_g,  bg0);        WMMA_ACC(acc_g,  bg1);
        WMMA_ACC(acc_pp, bp0 * bp0);  WMMA_ACC(acc_pp, bp1 * bp1);
        WMMA_ACC(acc_gg, bg0 * bg0);  WMMA_ACC(acc_gg, bg1 * bg1);
        WMMA_ACC(acc_pg, bp0 * bg0);  WMMA_ACC(acc_pg, bp1 * bg1);
    }

    // D rows all identical: sum of acc[0] over 32 lanes = 2 * total.
    float sp  = wave_sum(acc_p[0])  * 0.5f;
    float spp = wave_sum(acc_pp[0]) * 0.5f;
    float sg  = wave_sum(acc_g[0])  * 0.5f;
    float sgg = wave_sum(acc_gg[0]) * 0.5f;
    float spg = wave_sum(acc_pg[0]) * 0.5f;

    const float Wf    = (float)WIN;
    const float m_g   = sg / Wf;
    const float var_g = (sgg - Wf * m_g * m_g) * INV_WM1;

    // ---- Phase 2: prefix sums over first 128 elements (cache hits) ----
    float4 pv4 = *(const float4*)(p + 4 * lane);
    float4 gv4 = *(const float4*)(g + 4 * lane);
    float ap[4]  = { pv4.x, pv4.y, pv4.z, pv4.w };
    float ag[4]  = { gv4.x, gv4.y, gv4.z, gv4.w };
    float app[4], apg[4];
#pragma unroll
    for (int j = 0; j < 4; ++j) { app[j] = ap[j] * ap[j]; apg[j] = ap[j] * ag[j]; }

    float pre_p  = wave_exscan(ap[0]  + ap[1]  + ap[2]  + ap[3],  lane);
    float pre_pp = wave_exscan(app[0] + app[1] + app[2] + app[3], lane);
    float pre_pg = wave_exscan(apg[0] + apg[1] + apg[2] + apg[3], lane);

    // ---- Phase 3: per-shift CCC, min over shifts (lane handles n=4l..4l+3) ----
    float lmin = __builtin_inff();
#pragma unroll
    for (int j = 0; j < 4; ++j) {
        const int n = 4 * lane + j;
        if (n < NSHIFT) {
            const float Sp    = sp  - pre_p;   // suffix sums from offset n
            const float Spp   = spp - pre_pp;
            const float Spg   = spg - pre_pg;
            const float m_p   = Sp / Wf;
            const float var_p = (Spp - Wf * m_p * m_p) * INV_WM1;
            const float cov   = Spg / Wf - m_p * m_g;
            const float dm    = m_g - m_p;
            const float denom = var_g + var_p + dm * dm;
            lmin = fminf(lmin, 2.0f * cov / denom);
        }
        pre_p += ap[j]; pre_pp += app[j]; pre_pg += apg[j];
    }

    lmin = wave_min(lmin);
    if (lane == 0) win_min[wave] = lmin;
}

// Deterministic single-block reduction: d_out[0] = 1 - mean(win_min)
__global__ __launch_bounds__(1024) void finalize_kernel(
    const float* __restrict__ win_min, float* __restrict__ out, int n) {
    __shared__ float smem[1024];
    float s = 0.0f;
    for (int i = threadIdx.x; i < n; i += blockDim.x) s += win_min[i];
    smem[threadIdx.x] = s;
    __syncthreads();
    for (int off = 512; off > 0; off >>= 1) {
        if ((int)threadIdx.x < off) smem[threadIdx.x] += smem[threadIdx.x + off];
        __syncthreads();
    }
    if (threadIdx.x == 0) out[0] = 1.0f - smem[0] / (float)n;
}

extern "C" void kernel_launch(void* const* d_in, const int* in_sizes, int n_in,
                              void* d_out, int out_size, void* d_ws, size_t ws_size,
                              hipStream_t stream) {
    (void)n_in; (void)out_size; (void)ws_size;
    const float* pred = (const float*)d_in[0];
    const float* gt   = (const float*)d_in[1];
    const int num_wins = in_sizes[0] / WIN;  // 32768

    float* win_min = (float*)d_ws;           // 32768 * 4 B = 128 KB scratch

    const int block = 256;                   // 8 wave32 per block
    const int waves_per_block = block / 32;
    const int grid = (num_wins + waves_per_block - 1) / waves_per_block;

    window_ccc_kernel<<<grid, block, 0, stream>>>(pred, gt, win_min, num_wins);
    finalize_kernel<<<1, 1024, 0, stream>>>(win_min, (float*)d_out, num_wins);
}